// DAGNNConv_78125455114421
// MI455X (gfx1250) — compile-verified
//
#include <hip/hip_runtime.h>

#define FD 128    // feature dimension (fixed by the reference)
#define TPAD 132  // padded LDS row stride (floats): conflict-free ds_load_b64
#define NPB 64    // nodes per combine block (4 waves x 16 nodes)

typedef __attribute__((ext_vector_type(2))) float v2f;
typedef __attribute__((ext_vector_type(8))) float v8f;

// ---------------------------------------------------------------------------
// Degree histogram: in_deg[dst] += 1, out_deg[src] += 1  (float atomics)
// ---------------------------------------------------------------------------
__global__ void dagnn_deg_kernel(const int* __restrict__ src,
                                 const int* __restrict__ dst,
                                 float* __restrict__ in_deg,
                                 float* __restrict__ out_deg,
                                 int E) {
  int e = blockIdx.x * blockDim.x + threadIdx.x;
  if (e < E) {
    atomicAdd(&out_deg[src[e]], 1.0f);
    atomicAdd(&in_deg[dst[e]], 1.0f);
  }
}

// ---------------------------------------------------------------------------
// In-place: deg -> clip(deg,1)^-0.5
// ---------------------------------------------------------------------------
__global__ void dagnn_norm_kernel(float* __restrict__ in_deg,
                                  float* __restrict__ out_deg,
                                  int N) {
  int i = blockIdx.x * blockDim.x + threadIdx.x;
  if (i < N) {
    in_deg[i]  = rsqrtf(fmaxf(in_deg[i],  1.0f));  // becomes dst_norm
    out_deg[i] = rsqrtf(fmaxf(out_deg[i], 1.0f));  // becomes src_norm
  }
}

// ---------------------------------------------------------------------------
// One wave per edge: acc[dst,:] += h[src,:] * src_norm[src]
// Each lane moves one float4 (32 lanes * 4 = 128 floats). Working set
// (h: 51MB, acc: 51MB, edges: 25.6MB) is L2-resident on MI455X (192MB L2).
// ---------------------------------------------------------------------------
__global__ void dagnn_edge_kernel(const int* __restrict__ src,
                                  const int* __restrict__ dst,
                                  const float* __restrict__ h,
                                  const float* __restrict__ src_norm,
                                  float* __restrict__ acc,
                                  int E) {
  int e    = blockIdx.x * 8 + (threadIdx.x >> 5);
  int lane = threadIdx.x & 31;
  if (e >= E) return;
  int sN = src[e];
  int dN = dst[e];
  float sn = src_norm[sN];
  const float4* hp = (const float4*)(h + (size_t)sN * FD);
  float4 v = hp[lane];
  float* ap = acc + (size_t)dN * FD + lane * 4;
  atomicAdd(ap + 0, v.x * sn);
  atomicAdd(ap + 1, v.y * sn);
  atomicAdd(ap + 2, v.z * sn);
  atomicAdd(ap + 3, v.w * sn);
}

// ---------------------------------------------------------------------------
// Fused readout step for one propagation level k (64 nodes per block,
// 128 threads = 4 waves, each wave owns a 16-node WMMA chain):
//   1. stage 64x128 tile into padded LDS; if SCALE: multiply by dst_norm and
//      write back to h_out (the scaled tile is h_k, next propagation input)
//   2. each wave computes its 16 logits h_k . s with 32 chained
//      V_WMMA_F32_16X16X4_F32 ops. B carries s broadcast to ALL 16 columns,
//      so the fragment build is branch-free (no EXEC save/restore) and every
//      column of D equals the row's logit.
//   3. out[node,:] += sigmoid(logit[node]) * h_k[node,:]
// Exploits out = sum_k sigmoid(h_k.s) h_k being separable over k, so H
// (563MB) is never materialized.
// ---------------------------------------------------------------------------
template <int SCALE>
__global__ __launch_bounds__(128)
void dagnn_combine_kernel(const float* __restrict__ h_in,
                          const float* __restrict__ dst_norm,
                          const float* __restrict__ s,
                          float* __restrict__ h_out,
                          float* __restrict__ out,
                          int N) {
  __shared__ float tile[NPB * TPAD];
  __shared__ float s_lds[FD];
  __shared__ float logits[NPB];
  const int nodeBase = blockIdx.x * NPB;
  const int tid = threadIdx.x;  // 128 threads

  if (tid < FD) s_lds[tid] = s[tid];

  // Stage tile (coalesced float4), optionally scale + write back.
  for (int i = tid; i < NPB * (FD / 4); i += 128) {
    int node = i >> 5;        // FD/4 = 32 float4 per row
    int c4   = i & 31;
    int gnode = nodeBase + node;
    float4 v = {0.f, 0.f, 0.f, 0.f};
    if (gnode < N) {
      v = ((const float4*)(h_in + (size_t)gnode * FD))[c4];
      if (SCALE) {
        float dn = dst_norm[gnode];
        v.x *= dn; v.y *= dn; v.z *= dn; v.w *= dn;
        ((float4*)(h_out + (size_t)gnode * FD))[c4] = v;
      }
    }
    *(float4*)&tile[node * TPAD + c4 * 4] = v;
  }
  __syncthreads();

  // Every wave: 16 logits via chained f32 WMMA, K swept in 32 steps.
  {
    const int wave = tid >> 5;
    const int lane = tid & 31;
    const int m    = lane & 15;        // matrix row (node within sub-tile)
    const int half = lane >> 4;
    const int kb   = half * 2;         // K sub-offset per ISA 16x4 A-layout
    const float* arow = &tile[(wave * 16 + m) * TPAD + kb];
    const float* brow = &s_lds[kb];
    v8f c = {};
#pragma unroll
    for (int t = 0; t < FD / 4; ++t) {
      v2f a;                            // A[m][4t+kb], A[m][4t+kb+1]
      a.x = arow[4 * t];
      a.y = arow[4 * t + 1];
      v2f b;                            // B[k][n] = s[k] for every column n
      b.x = brow[4 * t];
      b.y = brow[4 * t + 1];
      c = __builtin_amdgcn_wmma_f32_16x16x4_f32(
          /*neg_a=*/false, a, /*neg_b=*/false, b,
          /*c_mod=*/(short)0, c, /*reuse_a=*/false, /*reuse_b=*/false);
    }
    // All columns equal; lanes with n==0 (m==0) carry rows half*8 .. half*8+7.
    if (m == 0) {
      int base = wave * 16 + half * 8;
      for (int r = 0; r < 8; ++r) logits[base + r] = c[r];
    }
  }
  __syncthreads();

  // out += sigmoid(logit) * h_k
  for (int i = tid; i < NPB * (FD / 4); i += 128) {
    int node = i >> 5;
    int c4   = i & 31;
    int gnode = nodeBase + node;
    if (gnode >= N) continue;
    float w = 1.0f / (1.0f + __expf(-logits[node]));
    float4 v = *(const float4*)&tile[node * TPAD + c4 * 4];
    float4* op = (float4*)(out + (size_t)gnode * FD);
    float4 o = op[c4];
    o.x += w * v.x; o.y += w * v.y; o.z += w * v.z; o.w += w * v.w;
    op[c4] = o;
  }
}

// ---------------------------------------------------------------------------
// Host orchestration (graph-capture safe: only async ops on `stream`).
// Inputs: d_in[0]=feats (N*128 f32), d_in[1]=s (128 f32),
//         d_in[2]=src (E i32), d_in[3]=dst (E i32)
// Workspace layout (floats): [in_deg(N) | out_deg(N) | bufA(N*128) | bufB(N*128)]
//   ~103 MB total for N=100000.
// ---------------------------------------------------------------------------
extern "C" void kernel_launch(void* const* d_in, const int* in_sizes, int n_in,
                              void* d_out, int out_size, void* d_ws, size_t ws_size,
                              hipStream_t stream) {
  const float* feats = (const float*)d_in[0];
  const float* s     = (const float*)d_in[1];
  const int*   src   = (const int*)d_in[2];
  const int*   dst   = (const int*)d_in[3];

  const int N = in_sizes[0] / FD;
  const int E = in_sizes[2];
  const size_t nd = (size_t)N * FD;

  float* ws      = (float*)d_ws;
  float* in_deg  = ws;           // becomes dst_norm
  float* out_deg = ws + N;       // becomes src_norm
  float* bufA    = ws + 2 * (size_t)N;
  float* bufB    = bufA + nd;
  float* out     = (float*)d_out;

  hipMemsetAsync(ws, 0, (size_t)2 * N * sizeof(float), stream);
  hipMemsetAsync(out, 0, nd * sizeof(float), stream);

  dagnn_deg_kernel<<<(E + 255) / 256, 256, 0, stream>>>(src, dst, in_deg, out_deg, E);
  dagnn_norm_kernel<<<(N + 255) / 256, 256, 0, stream>>>(in_deg, out_deg, N);

  const int combineBlocks = (N + NPB - 1) / NPB;

  // k = 0: level is feats itself (no dst_norm scaling, no write-back).
  dagnn_combine_kernel<0><<<combineBlocks, 128, 0, stream>>>(
      feats, nullptr, s, nullptr, out, N);

  const float* h_cur = feats;
  float* acc   = bufA;
  float* other = bufB;
  for (int k = 1; k <= 10; ++k) {
    hipMemsetAsync(acc, 0, nd * sizeof(float), stream);
    dagnn_edge_kernel<<<(E + 7) / 8, 256, 0, stream>>>(
        src, dst, h_cur, out_deg, acc, E);
    // In-place scale by dst_norm + logit + sigmoid-weighted accumulation.
    dagnn_combine_kernel<1><<<combineBlocks, 128, 0, stream>>>(
        acc, in_deg, s, acc, out, N);
    h_cur = acc;
    float* t = acc; acc = other; other = t;
  }
}